// GetPseudoMaskSLFCAMS_27530740367901
// MI455X (gfx1250) — compile-verified
//
#include <hip/hip_runtime.h>
#include <hip/hip_bf16.h>
#include <stdint.h>

// ---------------------------------------------------------------------------
// GetPseudoMaskSLFCAMS for MI455X (gfx1250)
//   kernel 1: per-image exact rank selection (3-pass radix histogram in LDS,
//             streamed through a deep double-buffered async-to-LDS pipeline)
//   kernel 2: ballot bit-packed compare + 3x3 binary dilation + seed write
// ---------------------------------------------------------------------------

#define USE_ASYNC_LDS 1
// b128 issues per thread per chunk. REP=4 -> 64 KB chunks, 2x64 KB LDS
// staging (CDNA5 allows up to 320 KB LDS per workgroup). REP=1 -> 16 KB
// chunks (proven fallback if the toolchain caps static LDS at 64 KB).
#define REP 4

typedef unsigned int uint32;

#define IMG_N 128
#define IMH 512
#define IMW 512
#define HW (IMH * IMW)          // 262144
#define RLO 26213u              // rank of k-th smallest (0-indexed), k=26214
#define RHI 235930u             // rank of (hw-k)-th smallest = smallest of top-k

#define CHUNK_F (4096 * REP)    // floats per chunk
#define NCHUNK (HW / CHUNK_F)

// Order-preserving float -> uint transform (ascending)
__device__ __forceinline__ uint32 ord_u32(float f) {
  uint32 u = __float_as_uint(f);
  return (u & 0x80000000u) ? ~u : (u | 0x80000000u);
}

// Low 32 bits of a generic pointer to __shared__ data == LDS byte address
__device__ __forceinline__ uint32 lds_lo32(const void* p) {
  return (uint32)(size_t)p;
}

__device__ __forceinline__ uint32 ballot32(bool p) {
#if defined(__AMDGCN__) && __has_builtin(__builtin_amdgcn_ballot_w32)
  return __builtin_amdgcn_ballot_w32(p);
#else
  return (uint32)__ballot(p);
#endif
}

#if USE_ASYNC_LDS
__device__ __forceinline__ void async_load_b128_to_lds(uint32 lds_byte_addr,
                                                       const void* sbase,
                                                       uint32 voffset_bytes) {
  // GLOBAL_LOAD_ASYNC_TO_LDS_B128 (GVS mode): vdst = LDS byte address VGPR,
  // vaddr = 32-bit byte offset, saddr = 64-bit SGPR base. Tracked by ASYNCcnt.
  asm volatile("global_load_async_to_lds_b128 %0, %1, %2"
               :: "v"(lds_byte_addr), "v"(voffset_bytes), "s"(sbase)
               : "memory");
}
template <int N>
__device__ __forceinline__ void wait_asynccnt() {
  asm volatile("s_wait_asynccnt %0" :: "i"(N) : "memory");
}
#endif

// ---------------------------------------------------------------------------
// Streaming helper: pipe one 1 MB image through LDS in NCHUNK chunks of
// CHUNK_F floats, double buffered. Each of 1024 threads issues REP async
// b128s per chunk (REP ASYNCcnt ticks per wave) and consumes 4*REP floats
// from the completed buffer. In-flight bytes per block = CHUNK_F*4.
// ---------------------------------------------------------------------------
template <typename F>
__device__ __forceinline__ void stream_image(const float* __restrict__ img,
                                             float (&buf)[2][CHUNK_F],
                                             int tid, F consume) {
#if USE_ASYNC_LDS
  const uint32 voff = (uint32)tid * 16u;
  const uint32 l0 = lds_lo32(&buf[0][0]) + voff;
  const uint32 l1 = lds_lo32(&buf[1][0]) + voff;
#pragma unroll
  for (int q = 0; q < REP; ++q)
    async_load_b128_to_lds(l0 + (uint32)q * 16384u, img,
                           voff + (uint32)q * 16384u);
  for (int c = 0; c < NCHUNK; ++c) {
    if (c + 1 < NCHUNK) {
      const char* gb = (const char*)img + (size_t)(c + 1) * (CHUNK_F * 4u);
      const uint32 ln = (c & 1) ? l0 : l1;
#pragma unroll
      for (int q = 0; q < REP; ++q)
        async_load_b128_to_lds(ln + (uint32)q * 16384u, gb,
                               voff + (uint32)q * 16384u);
      wait_asynccnt<REP>();   // chunk c complete; chunk c+1 (REP ops) in flight
    } else {
      wait_asynccnt<0>();
    }
    __syncthreads();          // publish async LDS writes across waves
    const float* b = buf[c & 1];
#pragma unroll
    for (int q = 0; q < REP; ++q) {
      consume(b[q * 4096 + 4 * tid + 0]);
      consume(b[q * 4096 + 4 * tid + 1]);
      consume(b[q * 4096 + 4 * tid + 2]);
      consume(b[q * 4096 + 4 * tid + 3]);
    }
    __syncthreads();          // all reads done before buffer reuse
  }
#else
  for (int c = 0; c < NCHUNK; ++c) {
    const float* g = img + (size_t)c * CHUNK_F;
    for (int q = 0; q < REP; ++q) {
      consume(g[q * 4096 + 4 * tid + 0]);
      consume(g[q * 4096 + 4 * tid + 1]);
      consume(g[q * 4096 + 4 * tid + 2]);
      consume(g[q * 4096 + 4 * tid + 3]);
    }
  }
  __syncthreads();
#endif
}

// Serial scan over n bins: first bin where cumulative count exceeds `rank`.
__device__ uint32 select_serial(const uint32* h, int n, uint32 rank,
                                uint32* cumBefore) {
  uint32 cum = 0;
  int i = 0;
  for (; i < n; ++i) {
    uint32 v = h[i];
    if (cum + v > rank) break;
    cum += v;
  }
  if (i == n) i = n - 1;   // defensive clamp (ranks are always in range)
  *cumBefore = cum;
  return (uint32)i;
}

// Two-level scan: supers of 32 bins, then 32 bins (thread-0 only, ~100 iters).
__device__ uint32 select_two_level(const uint32* hist, const uint32* super,
                                   int nsuper, uint32 rank, uint32* cumBefore) {
  uint32 c1, c2;
  uint32 s = select_serial(super, nsuper, rank, &c1);
  uint32 b = select_serial(hist + s * 32, 32, rank - c1, &c2);
  *cumBefore = c1 + c2;
  return s * 32 + b;
}

// ---------------------------------------------------------------------------
// Kernel 1: one block per image, 1024 threads.
// 3-pass radix select on ord_u32 bits 11/11/10 -> exact u_lo, u_hi thresholds.
// LDS: 16 KB hist + 2*CHUNK_F*4 staging (+aux).
// ---------------------------------------------------------------------------
__global__ __launch_bounds__(1024, 1)
void hist_select_kernel(const float* __restrict__ x, uint32* __restrict__ thr) {
  __shared__ uint32 sHist[4096];       // pass A: 2048; pass B: 2x2048; C: 2x1024
  __shared__ float  sBuf[2][CHUNK_F];  // async staging
  __shared__ uint32 sSuper[64];
  __shared__ uint32 sSel[4];

  const int tid = threadIdx.x;
  const float* img = x + (size_t)blockIdx.x * HW;

  // ---- pass A: bits [31:21] (2048 bins) -------------------------------
  for (int i = tid; i < 2048; i += 1024) sHist[i] = 0u;
  __syncthreads();
  stream_image(img, sBuf, tid, [&](float f) {
    atomicAdd(&sHist[ord_u32(f) >> 21], 1u);
  });
  if (tid < 64) {
    uint32 s = 0;
    for (int j = 0; j < 32; ++j) s += sHist[tid * 32 + j];
    sSuper[tid] = s;
  }
  __syncthreads();
  if (tid == 0) {
    uint32 c;
    sSel[0] = select_two_level(sHist, sSuper, 64, RLO, &c); sSel[1] = c;
    sSel[2] = select_two_level(sHist, sSuper, 64, RHI, &c); sSel[3] = c;
  }
  __syncthreads();
  const uint32 binLo1 = sSel[0], cumLo1 = sSel[1];
  const uint32 binHi1 = sSel[2], cumHi1 = sSel[3];
  __syncthreads();

  // ---- pass B: bits [20:10] within selected top bins (2 x 2048 bins) --
  for (int i = tid; i < 4096; i += 1024) sHist[i] = 0u;
  __syncthreads();
  stream_image(img, sBuf, tid, [&](float f) {
    uint32 u = ord_u32(f);
    uint32 t = u >> 21;
    uint32 mid = (u >> 10) & 0x7FFu;
    if (t == binLo1) atomicAdd(&sHist[mid], 1u);
    if (t == binHi1) atomicAdd(&sHist[2048u + mid], 1u);
  });
  if (tid < 64) {
    uint32 s = 0;
    for (int j = 0; j < 32; ++j) s += sHist[tid * 32 + j];
    sSuper[tid] = s;
  }
  __syncthreads();
  if (tid == 0) {
    uint32 c;
    sSel[0] = select_two_level(sHist, sSuper, 64, RLO - cumLo1, &c);
    sSel[1] = c;
  }
  __syncthreads();
  if (tid < 64) {
    uint32 s = 0;
    for (int j = 0; j < 32; ++j) s += sHist[2048 + tid * 32 + j];
    sSuper[tid] = s;
  }
  __syncthreads();
  if (tid == 0) {
    uint32 c;
    sSel[2] = select_two_level(sHist + 2048, sSuper, 64, RHI - cumHi1, &c);
    sSel[3] = c;
  }
  __syncthreads();
  const uint32 binLo2 = sSel[0], cumLo2 = sSel[1];
  const uint32 binHi2 = sSel[2], cumHi2 = sSel[3];
  const uint32 pfxLo = (binLo1 << 11) | binLo2;   // 22-bit prefix
  const uint32 pfxHi = (binHi1 << 11) | binHi2;
  __syncthreads();

  // ---- pass C: bits [9:0] within selected prefixes (2 x 1024 bins) ----
  for (int i = tid; i < 2048; i += 1024) sHist[i] = 0u;
  __syncthreads();
  stream_image(img, sBuf, tid, [&](float f) {
    uint32 u = ord_u32(f);
    uint32 p = u >> 10;
    if (p == pfxLo) atomicAdd(&sHist[u & 0x3FFu], 1u);
    if (p == pfxHi) atomicAdd(&sHist[1024u + (u & 0x3FFu)], 1u);
  });
  if (tid < 64) {   // 2 regions x 32 supers of 32 bins
    uint32 s = 0;
    int base = (tid < 32) ? (tid * 32) : (1024 + (tid - 32) * 32);
    for (int j = 0; j < 32; ++j) s += sHist[base + j];
    sSuper[tid] = s;
  }
  __syncthreads();
  if (tid == 0) {
    uint32 c;
    uint32 lowLo = select_two_level(sHist, sSuper, 32,
                                    RLO - cumLo1 - cumLo2, &c);
    uint32 lowHi = select_two_level(sHist + 1024, sSuper + 32, 32,
                                    RHI - cumHi1 - cumHi2, &c);
    thr[blockIdx.x * 2 + 0] = (pfxLo << 10) | lowLo;   // u_lo (rank k-1)
    thr[blockIdx.x * 2 + 1] = (pfxHi << 10) | lowHi;   // u_hi (rank hw-k)
  }
}

// ---------------------------------------------------------------------------
// Kernel 2: 8 bands of 64 rows per image. Ballot-packed compares, bitwise 3x3
// dilation in LDS, overlap removal, coalesced int32 seed writes.
// ---------------------------------------------------------------------------
__global__ __launch_bounds__(256)
void seed_kernel(const float* __restrict__ x, const uint32* __restrict__ thr,
                 int* __restrict__ out) {
  __shared__ uint32 fgRaw[66][16];   // rows r0-1 .. r0+64, 16 words of 32 cols
  __shared__ uint32 bgRaw[66][16];
  __shared__ uint32 fgD[64][16];
  __shared__ uint32 bgD[64][16];

  const int tid  = threadIdx.x;
  const int img  = blockIdx.x >> 3;
  const int band = blockIdx.x & 7;
  const int r0   = band * 64;
  const uint32 uLo = thr[img * 2 + 0];
  const uint32 uHi = thr[img * 2 + 1];
  const float* base = x + (size_t)img * HW;
  const int wave = tid >> 5, lane = tid & 31;

  // Phase 1: one wave per 32-pixel word; ballot packs compares into bits.
  for (int wi = wave; wi < 66 * 16; wi += 8) {
    int rr = wi >> 4, w = wi & 15;
    int row = r0 - 1 + rr;
    bool valid = (row >= 0) && (row < IMH);
    bool fgb = false, bgb = false;
    if (valid) {
      uint32 u = ord_u32(base[row * IMW + w * 32 + lane]);
      fgb = (u >= uHi);
      bgb = (u <= uLo);
    }
    uint32 fm = ballot32(fgb);
    uint32 bm = ballot32(bgb);
    if (lane == 0) { fgRaw[rr][w] = fm; bgRaw[rr][w] = bm; }
  }
  __syncthreads();

  // Phase 2: 3x3 dilation = horizontal smear (with cross-word carry) OR'ed
  // over three rows. Rows outside the image were stored as zero.
  for (int wi = tid; wi < 64 * 16; wi += 256) {
    int r = wi >> 4, w = wi & 15;
    uint32 fa = 0u, ba = 0u;
#pragma unroll
    for (int dr = 0; dr < 3; ++dr) {
      int rr = r + dr;   // raw row index (r0-1+rr == output row r + dr - 1)
      uint32 m = fgRaw[rr][w];
      uint32 p = (w > 0)  ? fgRaw[rr][w - 1] : 0u;
      uint32 n = (w < 15) ? fgRaw[rr][w + 1] : 0u;
      fa |= m | (m << 1) | (m >> 1) | (p >> 31) | (n << 31);
      m = bgRaw[rr][w];
      p = (w > 0)  ? bgRaw[rr][w - 1] : 0u;
      n = (w < 15) ? bgRaw[rr][w + 1] : 0u;
      ba |= m | (m << 1) | (m >> 1) | (p >> 31) | (n << 31);
    }
    fgD[r][w] = fa;
    bgD[r][w] = ba;
  }
  __syncthreads();

  // Phase 3: overlap removal + coalesced int32 seed writes.
  int* obase = out + (size_t)img * HW + (size_t)r0 * IMW;
  for (int p = tid; p < 64 * IMW; p += 256) {
    int r = p >> 9, c = p & 511;
    uint32 f = (fgD[r][c >> 5] >> (c & 31)) & 1u;
    uint32 g = (bgD[r][c >> 5] >> (c & 31)) & 1u;
    int seed = (f && !g) ? 1 : ((g && !f) ? 0 : -255);
    obase[p] = seed;
  }
}

// ---------------------------------------------------------------------------
extern "C" void kernel_launch(void* const* d_in, const int* in_sizes, int n_in,
                              void* d_out, int out_size, void* d_ws,
                              size_t ws_size, hipStream_t stream) {
  (void)in_sizes; (void)n_in; (void)out_size; (void)ws_size;
  const float* x = (const float*)d_in[0];
  uint32* thr = (uint32*)d_ws;               // 128 * 2 u32 thresholds
  int* out = (int*)d_out;                    // int32 seed map (ref dtype)

  hist_select_kernel<<<dim3(IMG_N), dim3(1024), 0, stream>>>(x, thr);
  seed_kernel<<<dim3(IMG_N * 8), dim3(256), 0, stream>>>(x, thr, out);
}